// BilinearInterpolation_13443247637074
// MI455X (gfx1250) — compile-verified
//
#include <hip/hip_runtime.h>
#include <stdint.h>

#define IMG_H 1024
#define IMG_W 1024

typedef float __attribute__((ext_vector_type(2))) vf2;
typedef float __attribute__((ext_vector_type(4))) vf4;

// One bilinear tap with zero-padding out-of-bounds semantics (branchless).
__device__ __forceinline__ float tap(const float* __restrict__ base, int y, int x) {
    // valid iff 0 <= x < W and 0 <= y < H
    bool valid = ((unsigned)x < (unsigned)IMG_W) & ((unsigned)y < (unsigned)IMG_H);
    int xc = x < 0 ? 0 : (x > IMG_W - 1 ? IMG_W - 1 : x);
    int yc = y < 0 ? 0 : (y > IMG_H - 1 ? IMG_H - 1 : y);
    float v = base[yc * IMG_W + xc];   // L2-resident gather (imgs fits in 192MB L2)
    return valid ? v : 0.0f;
}

__device__ __forceinline__ float sample_px(const float* __restrict__ base,
                                           int w, int h, float dx, float dy) {
    float fx = (float)w + dx;
    float fy = (float)h + dy;
    float x0f = floorf(fx);
    float y0f = floorf(fy);
    float wx = fx - x0f;
    float wy = fy - y0f;
    int x0 = (int)x0f;
    int y0 = (int)y0f;
    int x1 = x0 + 1;
    int y1 = y0 + 1;

    float v00 = tap(base, y0, x0);
    float v01 = tap(base, y0, x1);
    float v10 = tap(base, y1, x0);
    float v11 = tap(base, y1, x1);

    float omwx = 1.0f - wx;
    float omwy = 1.0f - wy;
    // Same product form as the reference for numerical fidelity.
    return v00 * omwx * omwy + v01 * wx * omwy + v10 * omwx * wy + v11 * wx * wy;
}

// Each thread handles 2 horizontally-adjacent pixels:
//  - 16B/lane dvfs load, staged Global->LDS via CDNA5 async DMA (non-temporal)
//  - 8 gather taps (L2 hits)
//  - 8B/lane non-temporal output store
__global__ __launch_bounds__(256) void BilinearResample_gfx1250_kernel(
    const float* __restrict__ imgs,
    const float* __restrict__ dvfs,   // viewed as float4 = 2 pixels of (dx,dy)
    float* __restrict__ out,          // viewed as float2 = 2 pixels
    int npairs)
{
    __shared__ vf4 s_dvf[256];

    int t = blockIdx.x * 256 + threadIdx.x;
    if (t >= npairs) return;   // never taken for this shape (2^24 pairs); keeps EXEC full

    // ---- CDNA5 async global->LDS staging of the dvfs stream (th:NT keeps it
    // out of the way of the L2-resident imgs working set) ----
    uint32_t lds_off = (uint32_t)(uintptr_t)(&s_dvf[threadIdx.x]);   // low 32b of flat = LDS offset
    uint64_t gaddr   = (uint64_t)(uintptr_t)(dvfs) + (uint64_t)t * 16u;
    asm volatile("global_load_async_to_lds_b128 %0, %1, off th:TH_LOAD_NT"
                 :: "v"(lds_off), "v"(gaddr) : "memory");
    asm volatile("s_wait_asynccnt 0x0" ::: "memory");

    vf4 d = s_dvf[threadIdx.x];   // (dx0, dy0, dx1, dy1)

    int p0 = t * 2;
    int w0 = p0 & (IMG_W - 1);
    int h  = (p0 >> 10) & (IMG_H - 1);
    int b  = p0 >> 20;
    const float* base = imgs + ((size_t)b << 20);

    float r0 = sample_px(base, w0,     h, d.x, d.y);
    float r1 = sample_px(base, w0 + 1, h, d.z, d.w);

    vf2 res;
    res.x = r0;
    res.y = r1;
    // Non-temporal 64-bit store: output is written once, never re-read.
    __builtin_nontemporal_store(res, (vf2*)(out + (size_t)p0));
}

extern "C" void kernel_launch(void* const* d_in, const int* in_sizes, int n_in,
                              void* d_out, int out_size, void* d_ws, size_t ws_size,
                              hipStream_t stream) {
    const float* imgs = (const float*)d_in[0];   // (B,H,W,1) fp32
    const float* dvfs = (const float*)d_in[1];   // (B,H,W,2) fp32
    float* out = (float*)d_out;                  // (B,H,W,1) fp32

    int npix   = in_sizes[0];        // B*H*W (C==1)
    int npairs = npix >> 1;          // 2 pixels per thread
    int blocks = (npairs + 255) / 256;

    BilinearResample_gfx1250_kernel<<<blocks, 256, 0, stream>>>(imgs, dvfs, out, npairs);
}